// SuperpixelPooling_50663434223993
// MI455X (gfx1250) — compile-verified
//
#include <hip/hip_runtime.h>

// ---------------- problem constants ----------------
#define BB 8
#define HH 512
#define WW 512
#define CC 64
#define SS 256
#define PP (HH * WW)                 // 262144 pixels per image
#define GROUPS_PER_IMAGE 64
#define PIX_PER_GROUP (PP / GROUPS_PER_IMAGE)   // 4096
#define CHUNK 32                     // pixels per TDM chunk
#define NCHUNK (PIX_PER_GROUP / CHUNK)          // 128
#define CHUNK_ELTS (CHUNK * CC)      // 2048 f32 = 8 KB per staging buffer

typedef unsigned int u32x4 __attribute__((ext_vector_type(4)));
typedef int          i32x4 __attribute__((ext_vector_type(4)));
typedef int          i32x8 __attribute__((ext_vector_type(8)));

// Low 32 bits of a generic pointer to __shared__ data == LDS byte offset
// (LDS aperture: high dword selects aperture, addr[31:0] is the LDS address).
__device__ __forceinline__ unsigned lds_byte_off(const void* p) {
    return (unsigned)(uintptr_t)p;
}

// Issue one 1-D TDM copy: nelts f32 elements, global -> LDS.
// D# layout per CDNA5 ISA sec 8.3/8.4:
//   group0: [1:0]=count=1, [63:32]=lds_addr, [120:64]=global_addr, [127:126]=type=2
//   group1: data_size=2 (4B) at [17:16]; tensor_dim0=[79:48]; tensor_dim1=[111:80]=1;
//           tile_dim0=[127:112]; tile_dim1=[143:128]=1; tensor_dim0_stride=[207:160]
__device__ __forceinline__ void tdm_load_1d(unsigned lds_addr,
                                            const void* gptr,
                                            unsigned nelts) {
    unsigned long long ga = (unsigned long long)(uintptr_t)gptr;
    u32x4 g0;
    g0.x = 1u;                                            // count=1, no gather
    g0.y = lds_addr;                                      // LDS dest (bytes)
    g0.z = (unsigned)(ga & 0xFFFFFFFFu);                  // global_addr[31:0]
    g0.w = (unsigned)((ga >> 32) & 0x1FFFFFFu) | (2u << 30); // addr[56:32] | type=2

    i32x8 g1;
    g1[0] = (int)(2u << 16);                              // data_size = 2 -> 4 bytes
    g1[1] = (int)((nelts & 0xFFFFu) << 16);               // tensor_dim0 lo16
    g1[2] = (int)(((nelts >> 16) & 0xFFFFu) | (1u << 16));// tensor_dim0 hi | tensor_dim1=1
    g1[3] = (int)((nelts & 0xFFFFu) << 16);               // tensor_dim1 hi=0 | tile_dim0
    g1[4] = 1;                                            // tile_dim1=1, tile_dim2=0
    g1[5] = (int)nelts;                                   // tensor_dim0_stride lo32
    g1[6] = 0;                                            // stride hi | dim1_stride lo
    g1[7] = 0;                                            // dim1_stride hi

    i32x4 z4 = {0, 0, 0, 0};
#if __clang_major__ >= 23
    i32x8 z8 = {0, 0, 0, 0, 0, 0, 0, 0};
    __builtin_amdgcn_tensor_load_to_lds(g0, g1, z4, z4, z8, 0);
#else
    __builtin_amdgcn_tensor_load_to_lds(g0, g1, z4, z4, 0);
#endif
}

// ---------------- kernel 1: TDM-streamed binning ----------------
__global__ __launch_bounds__(256)
void SuperpixelPooling_pool_kernel(const float* __restrict__ feat,
                                   const int* __restrict__ spx,
                                   float* __restrict__ out_sums,
                                   float* __restrict__ counts) {
    __shared__ float bins[SS * CC];              // 64 KB
    __shared__ float cnt[SS];                    // 1 KB
    __shared__ float stage[2][CHUNK_ELTS];       // 2 x 8 KB double buffer

    const int tid = threadIdx.x;
    const int b   = blockIdx.x / GROUPS_PER_IMAGE;
    const int grp = blockIdx.x % GROUPS_PER_IMAGE;
    const size_t pixBase = (size_t)b * PP + (size_t)grp * PIX_PER_GROUP;
    const float* featBase = feat + pixBase * CC;
    const int*   labBase  = spx + pixBase;

    // Kick the first DMA immediately (EXEC is ignored by tensor ops, so the
    // tid==0 guard yields exactly one TDM issue per workgroup). It overlaps
    // with the 64 KB LDS bin-zeroing below.
    if (tid == 0)
        tdm_load_1d(lds_byte_off(&stage[0][0]), featBase, CHUNK_ELTS);

    for (int i = tid; i < SS * CC; i += 256) bins[i] = 0.0f;
    cnt[tid] = 0.0f;                             // blockDim==256==SS
    __syncthreads();

    const int pair = tid & 31;                   // channel pair 0..31
    const int slot = tid >> 5;                   // pixel slot 0..7

    for (int c = 0; c < NCHUNK; ++c) {
        if (tid == 0) {
            if (c + 1 < NCHUNK) {
                tdm_load_1d(lds_byte_off(&stage[(c + 1) & 1][0]),
                            featBase + (size_t)(c + 1) * CHUNK_ELTS, CHUNK_ELTS);
                // tensor ops of one wave complete in order: cnt<=1 means chunk c landed
                __builtin_amdgcn_s_wait_tensorcnt(1);
            } else {
                __builtin_amdgcn_s_wait_tensorcnt(0);
            }
        }
        __syncthreads();                         // publish chunk c to all waves

        const float2* st = (const float2*)&stage[c & 1][0];
#pragma unroll
        for (int sub = 0; sub < CHUNK / 8; ++sub) {
            const int pix = slot + sub * 8;      // 0..CHUNK-1
            const int lab = labBase[c * CHUNK + pix];
            const float2 v = st[pix * 32 + pair];      // ds_load_b64, conflict-free
            atomicAdd(&bins[lab * CC + 2 * pair],     v.x);  // ds_add_f32
            atomicAdd(&bins[lab * CC + 2 * pair + 1], v.y);  // ds_add_f32
            if (pair == 0) atomicAdd(&cnt[lab], 1.0f);
        }
        __syncthreads();                         // buffer (c&1) safe to refill
    }

    // Flush per-group partials (output is 512 KB -> L2-resident atomics).
    float* outB = out_sums + (size_t)b * SS * CC;
    for (int i = tid; i < SS * CC; i += 256) {
        const float v = bins[i];
        if (v != 0.0f) atomicAdd(&outB[i], v);   // global_atomic_add_f32
    }
    const float cv = cnt[tid];
    if (cv != 0.0f) atomicAdd(&counts[b * SS + tid], cv);
}

// ---------------- kernel 2: zero accumulators ----------------
__global__ __launch_bounds__(256)
void SuperpixelPooling_zero_kernel(float* __restrict__ out,
                                   float* __restrict__ counts) {
    int i = blockIdx.x * blockDim.x + threadIdx.x;
    if (i < BB * SS * CC) out[i] = 0.0f;
    if (i < BB * SS) counts[i] = 0.0f;
}

// ---------------- kernel 3: divide by counts ----------------
__global__ __launch_bounds__(256)
void SuperpixelPooling_finalize_kernel(float* __restrict__ out,
                                       const float* __restrict__ counts) {
    int i = blockIdx.x * blockDim.x + threadIdx.x;
    if (i < BB * SS * CC) {
        const float c = counts[i >> 6];          // i / CC
        out[i] = out[i] / fmaxf(c, 1.0f);
    }
}

extern "C" void kernel_launch(void* const* d_in, const int* in_sizes, int n_in,
                              void* d_out, int out_size, void* d_ws, size_t ws_size,
                              hipStream_t stream) {
    const float* feat = (const float*)d_in[0];   // [B,H,W,C] f32
    const int*   spx  = (const int*)d_in[1];     // [B,H,W] i32
    float* out    = (float*)d_out;               // [B,S,C] f32
    float* counts = (float*)d_ws;                // B*S f32 scratch (8 KB)

    const int n = BB * SS * CC;
    SuperpixelPooling_zero_kernel<<<(n + 255) / 256, 256, 0, stream>>>(out, counts);
    SuperpixelPooling_pool_kernel<<<BB * GROUPS_PER_IMAGE, 256, 0, stream>>>(
        feat, spx, out, counts);
    SuperpixelPooling_finalize_kernel<<<(n + 255) / 256, 256, 0, stream>>>(out, counts);
}